// CONCATNet_7447473291796
// MI455X (gfx1250) — compile-verified
//
#include <hip/hip_runtime.h>

typedef __attribute__((ext_vector_type(2))) float v2f;
typedef __attribute__((ext_vector_type(8))) float v8f;

constexpr int Bc = 128, Nw = 4096, Sc = 32, Pc = 64, Dc = 128;
constexpr float NORM_INV = 1.0f / 300.0f;

constexpr int KA  = 256;   // stage|wafer K (dyn folded into rank-1 epilogue)
constexpr int LDA = 260;   // padded LDS row stride (floats): 260 % 64 == 4 -> conflict-free A loads
constexpr int KB  = 384;   // loc|wafer|next_stage K for robot arms
constexpr int LDB = 388;   // 388 % 64 == 4 -> conflict-free

// ---------------------------------------------------------------------------
// Kernel A: pm_emb = [stage|wafer]@W_concat[0:256] + remain_prs (x) w3
// grid = B*2 blocks; each block does 32 PM rows of one batch; 8 waves.
// Each wave owns one N tile and accumulates BOTH M tiles in one K sweep, so
// every B fragment load feeds two v_wmma_f32_16x16x4_f32 ops.
// ---------------------------------------------------------------------------
__global__ __launch_bounds__(256)
void pm_emb_kernel(const float* __restrict__ enc_row,
                   const float* __restrict__ enc_col,
                   const float* __restrict__ clockv,
                   const float* __restrict__ pet,
                   const float* __restrict__ W_dyn,
                   const float* __restrict__ W_concat,
                   const int*   __restrict__ lhw,
                   const int*   __restrict__ lstage,
                   float*       __restrict__ out)
{
    __shared__ float Xs[32 * LDA];   // 33,280 B
    __shared__ float w3[Dc];
    __shared__ float Rp[32];

    const int b    = blockIdx.x >> 1;
    const int half = blockIdx.x & 1;
    const int t    = threadIdx.x;

    // ---- stage gathered rows into LDS: row = [enc_col[stage] | enc_row[wafer]/0]
    {
        const int r     = t >> 3;        // 0..31 (8 threads per row)
        const int chunk = t & 7;         // 32 floats per chunk
        const int p     = half * 32 + r;
        const int col0  = chunk * 32;
        const float* src = nullptr;
        if (chunk < 4) {
            const int st = lstage[b * Pc + p] - 1;        // 0..S-1
            src = enc_col + ((size_t)b * Sc + st) * Dc + col0;
        } else {
            const int w = lhw[b * Pc + p];
            if (w >= 0 && w < Nw)
                src = enc_row + ((size_t)b * Nw + w) * Dc + (col0 - 128);
        }
        float* dst = &Xs[r * LDA + col0];
        if (src) {
            #pragma unroll
            for (int j = 0; j < 8; ++j)
                *(float4*)(dst + 4 * j) = *(const float4*)(src + 4 * j);
        } else {
            #pragma unroll
            for (int j = 0; j < 8; ++j)
                *(float4*)(dst + 4 * j) = make_float4(0.f, 0.f, 0.f, 0.f);
        }
    }
    // ---- w3[n] = sum_k W_dyn[0,k] * W_concat[256+k, n]; Rp = remain_prs
    if (t < Dc) {
        float acc = 0.f;
        #pragma unroll 4
        for (int k = 0; k < Dc; ++k)
            acc = fmaf(W_dyn[k], W_concat[(size_t)(256 + k) * Dc + t], acc);
        w3[t] = acc;
    } else if (t < 160) {
        const int r = t - 128;
        const int p = half * 32 + r;
        const float d = pet[b * Pc + p] - clockv[b];
        Rp[r] = (d > 0.f ? d : 0.f) * NORM_INV;
    }
    __syncthreads();

    // ---- WMMA GEMM: Xs[32,256] @ W_concat[0:256, 0:128]
    const int wave  = t >> 5;
    const int lane  = t & 31;
    const int l16   = lane & 15;
    const int khalf = (lane >> 4) << 1;          // 0 or 2 (A/B K split)
    const int n0    = wave * 16;                 // N tile per wave

    v8f acc0 = {};                               // M tile 0 (rows 0..15)
    v8f acc1 = {};                               // M tile 1 (rows 16..31)
    const float* wb = W_concat + n0 + l16;       // column n0+l16
    const float* a0 = &Xs[l16 * LDA + khalf];
    const float* a1 = &Xs[(16 + l16) * LDA + khalf];
    for (int k = 0; k < KA; k += 4) {
        // B frag (shared by both M tiles): v0 -> rows k/k+2, v1 -> k+1/k+3
        v2f bf;
        bf.x = wb[(size_t)(k + khalf)     * Dc];
        bf.y = wb[(size_t)(k + khalf + 1) * Dc];
        // A frags: lanes0-15 -> (K=k,k+1), lanes16-31 -> (K=k+2,k+3)
        v2f fa0 = *(const v2f*)(a0 + k);
        v2f fa1 = *(const v2f*)(a1 + k);
        acc0 = __builtin_amdgcn_wmma_f32_16x16x4_f32(
                   false, fa0, false, bf, (short)0, acc0, false, false);
        acc1 = __builtin_amdgcn_wmma_f32_16x16x4_f32(
                   false, fa1, false, bf, (short)0, acc1, false, false);
    }

    // rank-1 dyn epilogue + store.  C/D: vgpr r -> M=r (lanes<16) / M=r+8
    const float w3v   = w3[n0 + l16];
    const int   rhalf = (lane >> 4) << 3;        // 0 or 8
    #pragma unroll
    for (int r = 0; r < 8; ++r) {
        const int row0 = rhalf + r;              // M tile 0 row
        const float v0 = acc0[r] + Rp[row0] * w3v;
        out[((size_t)b * (Pc + 2) + half * 32 + row0) * Dc + n0 + l16] = v0;
    }
    #pragma unroll
    for (int r = 0; r < 8; ++r) {
        const int row1 = 16 + rhalf + r;         // M tile 1 row
        const float v1 = acc1[r] + Rp[row1] * w3v;
        out[((size_t)b * (Pc + 2) + half * 32 + row1) * Dc + n0 + l16] = v1;
    }
}

// ---------------------------------------------------------------------------
// Kernel B: robot arms.  256 arm rows total -> [256,384] @ W_robot[384,128].
// grid = 16 blocks; each block = one 16-row M tile; 8 waves = 8 N tiles.
// Reads pm_emb back from `out` (stream-ordered after kernel A).
// ---------------------------------------------------------------------------
__global__ __launch_bounds__(256)
void robot_kernel(const float* __restrict__ enc_row,
                  const float* __restrict__ enc_col,
                  const float* __restrict__ W_robot,
                  const int* __restrict__ a1loc, const int* __restrict__ a2loc,
                  const int* __restrict__ a1rec, const int* __restrict__ a2rec,
                  const int* __restrict__ a1ns,  const int* __restrict__ a2ns,
                  float* __restrict__ out)
{
    __shared__ float Xs[16 * LDB];   // 24,832 B
    const int t  = threadIdx.x;
    const int mt = blockIdx.x;       // 0..15

    // ---- stage gathered rows: [loc_emb | wafer | next_stage]
    {
        const int r     = t >> 4;               // 0..15
        const int fbase = (t & 15) * 6;         // 96 float4 per row, 6 per thread
        const int ar    = mt * 16 + r;
        const int b     = ar >> 1, arm = ar & 1;
        const int loc = (arm ? a2loc : a1loc)[b];
        const int rec = (arm ? a2rec : a1rec)[b];
        const int ns  = (arm ? a2ns  : a1ns )[b];
        #pragma unroll
        for (int j = 0; j < 6; ++j) {
            const int col = (fbase + j) * 4;    // 0..380
            const int seg = col >> 7, cin = col & 127;
            float4 v = make_float4(0.f, 0.f, 0.f, 0.f);
            if (seg == 0) {
                if (loc == Pc + 1)
                    v = make_float4(1.f, 1.f, 1.f, 1.f);
                else if (loc >= 1 && loc <= Pc)
                    v = *(const float4*)(out + ((size_t)b * (Pc + 2) + (loc - 1)) * Dc + cin);
            } else if (seg == 1) {
                if (rec >= 0 && rec < Nw)
                    v = *(const float4*)(enc_row + ((size_t)b * Nw + rec) * Dc + cin);
            } else {
                if (ns >= 1 && ns <= Sc)
                    v = *(const float4*)(enc_col + ((size_t)b * Sc + (ns - 1)) * Dc + cin);
            }
            *(float4*)&Xs[r * LDB + col] = v;
        }
    }
    __syncthreads();

    const int wave  = t >> 5;
    const int lane  = t & 31;
    const int l16   = lane & 15;
    const int khalf = (lane >> 4) << 1;
    const int n0    = wave * 16;

    v8f acc = {};
    const float* wb = W_robot + n0 + l16;
    const float* ap = &Xs[l16 * LDB + khalf];
    for (int k = 0; k < KB; k += 4) {
        v2f a = *(const v2f*)(ap + k);
        v2f bf;
        bf.x = wb[(size_t)(k + khalf)     * Dc];
        bf.y = wb[(size_t)(k + khalf + 1) * Dc];
        acc = __builtin_amdgcn_wmma_f32_16x16x4_f32(
                  false, a, false, bf, (short)0, acc, false, false);
    }
    #pragma unroll
    for (int r = 0; r < 8; ++r) {
        const int mrow = ((lane >> 4) << 3) + r;        // 0..15
        const int ar   = mt * 16 + mrow;
        const int b    = ar >> 1, arm = ar & 1;
        out[((size_t)b * (Pc + 2) + Pc + arm) * Dc + n0 + l16] = acc[r];
    }
}

// ---------------------------------------------------------------------------
extern "C" void kernel_launch(void* const* d_in, const int* in_sizes, int n_in,
                              void* d_out, int out_size, void* d_ws, size_t ws_size,
                              hipStream_t stream) {
    (void)in_sizes; (void)n_in; (void)out_size; (void)d_ws; (void)ws_size;
    const float* enc_row  = (const float*)d_in[0];
    const float* enc_col  = (const float*)d_in[1];
    const float* clockv   = (const float*)d_in[2];
    const float* pet      = (const float*)d_in[3];
    const float* W_dyn    = (const float*)d_in[4];
    const float* W_concat = (const float*)d_in[5];
    const float* W_robot  = (const float*)d_in[6];
    const int*   lhw      = (const int*)d_in[7];
    const int*   lstage   = (const int*)d_in[8];
    const int*   a1loc    = (const int*)d_in[9];
    const int*   a2loc    = (const int*)d_in[10];
    const int*   a1rec    = (const int*)d_in[11];
    const int*   a2rec    = (const int*)d_in[12];
    const int*   a1ns     = (const int*)d_in[13];
    const int*   a2ns     = (const int*)d_in[14];
    float* out = (float*)d_out;

    pm_emb_kernel<<<Bc * 2, 256, 0, stream>>>(enc_row, enc_col, clockv, pet,
                                              W_dyn, W_concat, lhw, lstage, out);
    robot_kernel<<<16, 256, 0, stream>>>(enc_row, enc_col, W_robot,
                                         a1loc, a2loc, a1rec, a2rec, a1ns, a2ns, out);
}